// ChainLoss_27075473834428
// MI455X (gfx1250) — compile-verified
//
#include <hip/hip_runtime.h>
#include <math.h>

#ifndef __has_builtin
#define __has_builtin(x) 0
#endif

#define LEAKY 0.1f

typedef float v2f __attribute__((ext_vector_type(2)));
typedef float v8f __attribute__((ext_vector_type(8)));

// ---------------------------------------------------------------------------
// gfx1250 async global->LDS path (ASYNCcnt).  Builtin signature (probe-
// verified in round 1/2): (int4* AS(1) global, int4* AS(3) LDS, imm offset,
// imm cpol).  Guarded: plain staging fallback if builtins are not declared.
// ---------------------------------------------------------------------------
#if __has_builtin(__builtin_amdgcn_global_load_async_to_lds_b128) && \
    __has_builtin(__builtin_amdgcn_s_wait_asynccnt)
#define HAVE_ASYNC_LDS 1
typedef int v4i __attribute__((vector_size(16)));
typedef __attribute__((address_space(1))) v4i v4i_glob;
typedef __attribute__((address_space(3))) v4i v4i_lds;
#else
#define HAVE_ASYNC_LDS 0
#endif

__device__ __forceinline__ float clip_exp(float v) {
  v = fminf(fmaxf(v, -30.0f), 30.0f);
  return __expf(v);  // v_exp_f32
}

// Stage n floats (n multiple of 4 on the async path; P=3000 ok) from global
// into LDS, cooperatively across the block.  Async version only *issues*;
// completion is awaited with s_wait_asynccnt at the consumer side.
__device__ __forceinline__ void row_to_lds(const float* __restrict__ g,
                                           float* l, int n, int tid, int NT) {
#if HAVE_ASYNC_LDS
  v4i_glob* gp = (v4i_glob*)g;  // C-style cast: const_cast + addrspacecast
  v4i_lds*  lp = (v4i_lds*)l;
  const int nv = n >> 2;        // 16-byte packets
  for (int i = tid; i < nv; i += NT)
    __builtin_amdgcn_global_load_async_to_lds_b128(gp + i, lp + i, 0, 0);
#else
  for (int i = tid; i < n; i += NT) l[i] = g[i];
#endif
}

// ---------------------------------------------------------------------------
// Block-wide sum for blockDim==1024.  Wave 0 folds 1024 partials to 32 lane
// accumulators, then a single V_WMMA_F32_16X16X4_F32 with A==ones performs the
// 32->16 cross-lane-half fold: D[m][n] = B[0][n] + B[2][n] = p(n) + p(n+16).
// EXEC is all-ones inside wave 0 (uniform guard), as WMMA requires.
// ---------------------------------------------------------------------------
__device__ __forceinline__ float block_sum_1024(float v, float* red, int tid) {
  red[tid] = v;
  __syncthreads();
  if (tid < 32) {
    float acc = 0.f;
#pragma unroll
    for (int k = 0; k < 32; ++k) acc += red[tid + (k << 5)];
    v2f a;  a[0] = 1.0f; a[1] = 1.0f;   // A (16x4) = ones
    v2f bm; bm[0] = acc; bm[1] = 0.0f;  // B rows 0/2 = partials, rows 1/3 = 0
    v8f c = {};
    v8f d = __builtin_amdgcn_wmma_f32_16x16x4_f32(
        false, a, false, bm, (short)0, c, false, false);
    if (tid < 16) red[tid] = d[0];      // pair sums p(n)+p(n+16)
  }
  __syncthreads();
  if (tid == 0) {
    float s = 0.f;
#pragma unroll
    for (int k = 0; k < 16; ++k) s += red[k];
    red[0] = s;
  }
  __syncthreads();
  float s = red[0];
  __syncthreads();  // red is reused by the caller
  return s;
}

__device__ __forceinline__ float block_sum_256(float v, float* red, int tid) {
  red[tid] = v;
  __syncthreads();
#pragma unroll
  for (int off = 128; off > 0; off >>= 1) {
    if (tid < off) red[tid] += red[tid + off];
    __syncthreads();
  }
  float s = red[0];
  __syncthreads();
  return s;
}

// One-time AoS repack of the denominator edge list so the hot loop does one
// global_load_b128 per edge instead of four global_load_b32s.
__global__ void pack_edges(const int* __restrict__ f, const int* __restrict__ t,
                           const int* __restrict__ p, const float* __restrict__ w,
                           int4* __restrict__ out, int E) {
  int e = blockIdx.x * blockDim.x + threadIdx.x;
  if (e < E) out[e] = make_int4(f[e], t[e], p[e], __float_as_int(w[e]));
}

// ---------------------------------------------------------------------------
// Denominator forward: one block per sequence, all HMM state in LDS.
// Per step: async-prefetch next obs row, exp current row in place, 40K-edge
// gather/scatter with ds_add_f32, one WMMA-assisted reduction, closed-form
// scale = tot * (1 + LEAKY * sum(init)).
// ---------------------------------------------------------------------------
template <bool PACKED>
__launch_bounds__(1024, 1)
__global__ void den_fwd(const float* __restrict__ x,
                        const int4* __restrict__ epack,
                        const int* __restrict__ efrom, const int* __restrict__ eto,
                        const int* __restrict__ epdf, const float* __restrict__ eprob,
                        const float* __restrict__ init, const float* __restrict__ fin,
                        float* __restrict__ out, int T, int P, int S, int E) {
  const int b = blockIdx.x, tid = threadIdx.x, NT = blockDim.x;
  __shared__ float alpha[2048];    // S <= 2048
  __shared__ float nalpha[2048];
  __shared__ float initL[2048];
  __shared__ float obs[2][3072];   // P <= 3072, double buffered
  __shared__ float red[1024];

  for (int i = tid; i < S; i += NT) {
    float v = init[i];
    initL[i] = v;
    alpha[i] = v;
  }
  const float* xb = x + (size_t)b * T * P;
  row_to_lds(xb, &obs[0][0], P, tid, NT);  // prefetch row 0 (async)

  float p0 = 0.f;
  for (int i = tid; i < S; i += NT) p0 += initL[i];
  const float init_sum = block_sum_1024(p0, red, tid);
  const float leaky_scale = 1.0f + LEAKY * init_sum;

  float logz = 0.f;
  for (int t = 0; t < T; ++t) {
    const int cur = t & 1, nxt = cur ^ 1;
#if HAVE_ASYNC_LDS
    __builtin_amdgcn_s_wait_asynccnt(0);   // raw row t landed in obs[cur]
#endif
    __syncthreads();
    if (t + 1 < T)                          // overlap HBM fetch of row t+1
      row_to_lds(xb + (size_t)(t + 1) * P, &obs[nxt][0], P, tid, NT);
    for (int i = tid; i < P; i += NT) obs[cur][i] = clip_exp(obs[cur][i]);
    for (int i = tid; i < S; i += NT) nalpha[i] = 0.f;
    __syncthreads();

    if (PACKED) {
      for (int e = tid; e < E; e += NT) {   // one b128 per edge, L2-resident
        int4 ed = epack[e];
        float c = alpha[ed.x] * __int_as_float(ed.w) * obs[cur][ed.z];
        atomicAdd(&nalpha[ed.y], c);        // ds_add_f32
      }
    } else {
      for (int e = tid; e < E; e += NT) {
        float c = alpha[efrom[e]] * eprob[e] * obs[cur][epdf[e]];
        atomicAdd(&nalpha[eto[e]], c);
      }
    }
    __syncthreads();

    float part = 0.f;
    for (int i = tid; i < S; i += NT) part += nalpha[i];
    const float tot = block_sum_1024(part, red, tid);
    const float scale = tot * leaky_scale;  // == sum(nalpha + LEAKY*tot*init)
    const float lt = LEAKY * tot, inv = 1.0f / scale;
    for (int i = tid; i < S; i += NT)
      alpha[i] = (nalpha[i] + lt * initL[i]) * inv;
    logz += __logf(scale);
    __syncthreads();
  }

  float part = 0.f;
  for (int i = tid; i < S; i += NT) part += alpha[i] * fin[i];
  const float dot = block_sum_1024(part, red, tid);
  if (tid == 0) out[b] = logz + __logf(dot);
}

// ---------------------------------------------------------------------------
// Numerator forward: tiny per-utterance graphs (S=100, E=400).  Only ~E of
// the P pdfs are touched per step, so gather x directly and exp inline.
// ---------------------------------------------------------------------------
__launch_bounds__(256, 4)
__global__ void num_fwd(const float* __restrict__ x,
                        const int* __restrict__ efrom, const int* __restrict__ eto,
                        const int* __restrict__ epdf, const float* __restrict__ eprob,
                        const float* __restrict__ init, const float* __restrict__ fin,
                        float* __restrict__ out, int T, int P, int S, int E) {
  const int b = blockIdx.x, tid = threadIdx.x, NT = blockDim.x;
  __shared__ float alpha[128], nalpha[128], initL[128], finalL[128];
  __shared__ float red[256];

  const int* bf = efrom + b * E;
  const int* bt = eto + b * E;
  const int* bp = epdf + b * E;
  const float* bw = eprob + b * E;
  const float* bi = init + b * S;
  const float* bfn = fin + b * S;

  if (tid < S) {
    float v = bi[tid];
    initL[tid] = v;
    alpha[tid] = v;
    finalL[tid] = bfn[tid];
  }
  const float init_sum = block_sum_256(tid < S ? bi[tid] : 0.f, red, tid);
  const float leaky_scale = 1.0f + LEAKY * init_sum;
  const float* xb = x + (size_t)b * T * P;

  float logz = 0.f;
  for (int t = 0; t < T; ++t) {
    if (tid < S) nalpha[tid] = 0.f;
    __syncthreads();
    const float* xr = xb + (size_t)t * P;
    for (int e = tid; e < E; e += NT) {
      float c = alpha[bf[e]] * bw[e] * clip_exp(xr[bp[e]]);
      atomicAdd(&nalpha[bt[e]], c);
    }
    __syncthreads();
    const float tot = block_sum_256(tid < S ? nalpha[tid] : 0.f, red, tid);
    const float scale = tot * leaky_scale;
    const float lt = LEAKY * tot, inv = 1.0f / scale;
    if (tid < S) alpha[tid] = (nalpha[tid] + lt * initL[tid]) * inv;
    logz += __logf(scale);
    __syncthreads();
  }
  const float dot =
      block_sum_256(tid < S ? alpha[tid] * finalL[tid] : 0.f, red, tid);
  if (tid == 0) out[b] = logz + __logf(dot);
}

__global__ void combine(const float* __restrict__ denL,
                        const float* __restrict__ numL, float* __restrict__ out,
                        int B, float invBT) {
  if (threadIdx.x == 0 && blockIdx.x == 0) {
    float d = 0.f, n = 0.f;
    for (int b = 0; b < B; ++b) {
      d += denL[b];
      n += numL[b];
    }
    out[0] = -(n - d) * invBT;
  }
}

// ---------------------------------------------------------------------------
// Inputs (setup_inputs order):
//  0 x[B,T,P] f32 | 1 den_from[E] | 2 den_to | 3 den_pdf | 4 den_prob f32
//  5 den_init[S] f32 | 6 den_final[S] f32
//  7 num_from[B,E'] | 8 num_to | 9 num_pdf | 10 num_prob f32
//  11 num_init[B,S'] f32 | 12 num_final[B,S'] f32
// Output: scalar f32.
// ---------------------------------------------------------------------------
extern "C" void kernel_launch(void* const* d_in, const int* in_sizes, int n_in,
                              void* d_out, int out_size, void* d_ws,
                              size_t ws_size, hipStream_t stream) {
  (void)n_in; (void)out_size;
  const float* x     = (const float*)d_in[0];
  const int*   dfrom = (const int*)d_in[1];
  const int*   dto   = (const int*)d_in[2];
  const int*   dpdf  = (const int*)d_in[3];
  const float* dprob = (const float*)d_in[4];
  const float* dinit = (const float*)d_in[5];
  const float* dfin  = (const float*)d_in[6];
  const int*   nfrom = (const int*)d_in[7];
  const int*   nto   = (const int*)d_in[8];
  const int*   npdf  = (const int*)d_in[9];
  const float* nprob = (const float*)d_in[10];
  const float* ninit = (const float*)d_in[11];
  const float* nfin  = (const float*)d_in[12];

  const int B = 32, T = 500;
  const int P = in_sizes[0] / (B * T);     // 3000
  const int E_den = in_sizes[1];           // 40000
  const int S_den = in_sizes[5];           // 2000
  const int E_num = in_sizes[7] / B;       // 400
  const int S_num = in_sizes[11] / B;      // 100

  float* denL = (float*)d_ws;                       // [B]
  float* numL = denL + B;                           // [B]
  int4*  epack = (int4*)((char*)d_ws + 256);        // [E_den], 16B aligned

  const size_t ws_need = 256 + (size_t)E_den * sizeof(int4);
  if (ws_size >= ws_need) {
    pack_edges<<<(E_den + 255) / 256, 256, 0, stream>>>(dfrom, dto, dpdf,
                                                        dprob, epack, E_den);
    den_fwd<true><<<B, 1024, 0, stream>>>(x, epack, dfrom, dto, dpdf, dprob,
                                          dinit, dfin, denL, T, P, S_den,
                                          E_den);
  } else {
    den_fwd<false><<<B, 1024, 0, stream>>>(x, nullptr, dfrom, dto, dpdf, dprob,
                                           dinit, dfin, denL, T, P, S_den,
                                           E_den);
  }
  num_fwd<<<B, 256, 0, stream>>>(x, nfrom, nto, npdf, nprob, ninit, nfin,
                                 numL, T, P, S_num, E_num);
  combine<<<1, 64, 0, stream>>>(denL, numL, (float*)d_out, B,
                                1.0f / (float)(B * T));
}